// NeuralStack_13022340842145
// MI455X (gfx1250) — compile-verified
//
#include <hip/hip_runtime.h>
#include <hip/hip_bf16.h>

#define SEQ_LEN 4096
#define WW      512
#define DD      64
#define CHUNK   64
#define NCHUNK  (SEQ_LEN / CHUNK)   // 64
#define NS_EPS  1e-6f

typedef __attribute__((ext_vector_type(2))) float v2f;
typedef __attribute__((ext_vector_type(8))) float v8f;

// ---------------------------------------------------------------------------
// Kernel A: sequential strength recursion (the only nonlinear part).
// One wave32; lane owns strength[lane] and strength[lane+32].
// Emits rw[t][d] = (tot>eps) ? strength[d]/(tot+eps) : 0  (read weights,
// with the tot<=eps zeroing folded in). Pure shfl scans — zero barriers.
// ---------------------------------------------------------------------------
__global__ __launch_bounds__(32) void ns_strength_scan(
    const float* __restrict__ actions,   // [SEQ_LEN][3]
    float* __restrict__ rw_out)          // [SEQ_LEN][64]
{
  const int lane = threadIdx.x;          // 0..31
  float s0 = 0.0f, s1 = 0.0f;            // strength[lane], strength[lane+32]
  for (int t = 0; t < SEQ_LEN; ++t) {
    const float push = actions[t * 3 + 0];
    const float pop  = actions[t * 3 + 1];
    const float noop = actions[t * 3 + 2];

    // inclusive prefix sums over 64 entries (two half-scans + splice)
    float p0 = s0, p1 = s1;
    #pragma unroll
    for (int off = 1; off < 32; off <<= 1) {
      float u0 = __shfl_up(p0, off);
      float u1 = __shfl_up(p1, off);
      if (lane >= off) { p0 += u0; p1 += u1; }
    }
    const float lowTot = __shfl(p0, 31);
    const float P0 = p0;
    const float P1 = p1 + lowTot;
    const float tot = __shfl(p1, 31) + lowTot;

    const float inv   = 1.0f / (tot + NS_EPS);
    const bool  valid = tot > NS_EPS;
    rw_out[t * 64 + lane]      = valid ? s0 * inv : 0.0f;
    rw_out[t * 64 + 32 + lane] = valid ? s1 * inv : 0.0f;

    // pop: reduction[i] = min(max(pop - sum_{j>i} s[j], 0), s[i])
    const float ca0 = tot - P0, ca1 = tot - P1;
    const float sp0 = s0 - fminf(fmaxf(pop - ca0, 0.0f), s0);
    const float sp1 = s1 - fminf(fmaxf(pop - ca1, 0.0f), s1);

    // push shift: [push; s[:-1]]
    const float s0top = __shfl(s0, 31);
    float sh0 = __shfl_up(s0, 1);
    float sh1 = __shfl_up(s1, 1);
    if (lane == 0) { sh0 = push; sh1 = s0top; }

    float n0 = noop * s0 + pop * sp0 + push * sh0;
    float n1 = noop * s1 + pop * sp1 + push * sh1;

    // renormalize if total exceeds depth
    float sum = n0 + n1;
    #pragma unroll
    for (int off = 16; off >= 1; off >>= 1) sum += __shfl_xor(sum, off);
    const float scale = (sum > (float)DD) ? ((float)DD / sum) : 1.0f;
    s0 = n0 * scale;
    s1 = n1 * scale;
  }
}

// ---------------------------------------------------------------------------
// Kernel B: per-chunk linearization. All step matrices are polynomials in the
// nilpotent shift S (S^64 == 0) and commute; products are lower-tri Toeplitz.
// One block per chunk; lane i owns polynomial P_i = p_i * M_{i+1..j-1} e0 in
// 64 registers (fully unrolled, static indices only).
// Outputs per chunk: R (read rows), L (strictly-lower local weights),
// M (Toeplitz chunk propagator, expanded), B (input->state).
// ---------------------------------------------------------------------------
__global__ __launch_bounds__(64) void ns_chunk_mats(
    const float* __restrict__ actions,
    const float* __restrict__ rw,        // [SEQ_LEN][64]
    float* __restrict__ Rg,              // [NCHUNK][64][64]
    float* __restrict__ Lg,              // [NCHUNK][64][64]
    float* __restrict__ Mg,              // [NCHUNK][64][64]
    float* __restrict__ Bg)              // [NCHUNK][64][64]
{
  const int k   = blockIdx.x;            // chunk index
  const int tid = threadIdx.x;           // 0..63
  __shared__ float m_sh[64];             // M_{0..j-1} e0 coefficients
  __shared__ float rw_sh[64];

  float P[64];
  #pragma unroll
  for (int d = 0; d < 64; ++d) P[d] = 0.0f;
  m_sh[tid] = (tid == 0) ? 1.0f : 0.0f;
  __syncthreads();

  for (int j = 0; j < CHUNK; ++j) {
    const int t = k * CHUNK + j;
    const float push = actions[t * 3 + 0];
    const float pop  = actions[t * 3 + 1];
    const float kf   = actions[t * 3 + 2] + pop;   // noop + pop
    rw_sh[tid] = rw[t * 64 + tid];
    __syncthreads();

    // R[j][tid] = sum_{dd>=tid} rw[dd] * m[dd-tid]   (Toeplitz correlation)
    float racc = 0.0f;
    for (int dd = tid; dd < 64; ++dd) racc += rw_sh[dd] * m_sh[dd - tid];
    Rg[(k * 64 + j) * 64 + tid] = racc;

    // L[j][tid] = rw_j . P_tid  (zero for tid >= j since P still zero)
    float lacc = 0.0f;
    #pragma unroll
    for (int d = 0; d < 64; ++d) lacc += rw_sh[d] * P[d];
    Lg[(k * 64 + j) * 64 + tid] = lacc;

    // P_i <- (kf I + push S) P_i   (static register indices)
    #pragma unroll
    for (int d = 63; d >= 1; --d) P[d] = kf * P[d] + push * P[d - 1];
    P[0] = kf * P[0];
    if (tid == j) P[0] = push;     // inject this step's value coefficient

    // m <- (kf I + push S) m
    const float mown  = m_sh[tid];
    const float mprev = (tid > 0) ? m_sh[tid - 1] : 0.0f;
    __syncthreads();
    m_sh[tid] = kf * mown + push * mprev;
    __syncthreads();
  }

  // B[:,i] = final P_i  (p_i * M_{i+1..63} e0)
  #pragma unroll
  for (int d = 0; d < 64; ++d) Bg[(k * 64 + d) * 64 + tid] = P[d];
  // expand Toeplitz propagator: M[d][d'] = m[d-d'] for d >= d'
  for (int d = 0; d < 64; ++d)
    Mg[(k * 64 + d) * 64 + tid] = (d >= tid) ? m_sh[d - tid] : 0.0f;
}

// ---------------------------------------------------------------------------
// Kernel C: chunked scan + output via v_wmma_f32_16x16x4_f32.
//   O_k = R_k X_k + L_k V_k      (streamed to d_out)
//   X_{k+1} = M_k X_k + B_k V_k  (wave-private state in LDS, 64x16 f32)
// Each wave owns one 16-column slice; the k-loop is wave-local -> no barriers
// (per-wave DS ops are in order). 32 waves cover W=512.
// ---------------------------------------------------------------------------
__global__ __launch_bounds__(256) void ns_scan_gemm(
    const float* __restrict__ values,    // [SEQ_LEN][W]
    const float* __restrict__ Rg, const float* __restrict__ Lg,
    const float* __restrict__ Mg, const float* __restrict__ Bg,
    float* __restrict__ out)             // [SEQ_LEN][W]
{
  __shared__ float Xsh[8][64 * 16];      // per-wave state slice (32 KB)
  const int wid  = threadIdx.x >> 5;
  const int lane = threadIdx.x & 31;
  const int lo = lane & 15;
  const int hi = lane >> 4;
  const int cg = blockIdx.x * 8 + wid;   // column group 0..31
  const int col0 = cg * 16;
  float* XL = &Xsh[wid][0];

  for (int i = lane; i < 64 * 16; i += 32) XL[i] = 0.0f;   // X_0 = 0

  for (int k = 0; k < NCHUNK; ++k) {
    const float* Rk = Rg + k * 64 * 64;
    const float* Lk = Lg + k * 64 * 64;
    const float* Mk = Mg + k * 64 * 64;
    const float* Bk = Bg + k * 64 * 64;
    const float* Vk = values + (size_t)(k * CHUNK) * WW;
    if (k + 1 < NCHUNK)
      __builtin_prefetch(values + (size_t)((k + 1) * CHUNK) * WW + col0, 0, 1);

    v8f accO[4], accX[4];
    #pragma unroll
    for (int mt = 0; mt < 4; ++mt) {
      v8f o = {};
      v8f x = {};
      #pragma unroll
      for (int K0 = 0; K0 < 64; K0 += 4) {
        const int krow = K0 + 2 * hi;
        // B-operand fragments (4x16): state slice from LDS, values from HBM
        v2f bX, bV;
        bX.x = XL[krow * 16 + lo];
        bX.y = XL[(krow + 1) * 16 + lo];
        bV.x = Vk[(size_t)krow * WW + col0 + lo];
        bV.y = Vk[(size_t)(krow + 1) * WW + col0 + lo];
        // A-operand fragments (16x4) from the four chunk matrices
        const int arow = mt * 16 + lo;
        v2f aR, aL, aM, aB;
        aR.x = Rk[arow * 64 + krow]; aR.y = Rk[arow * 64 + krow + 1];
        aL.x = Lk[arow * 64 + krow]; aL.y = Lk[arow * 64 + krow + 1];
        aM.x = Mk[arow * 64 + krow]; aM.y = Mk[arow * 64 + krow + 1];
        aB.x = Bk[arow * 64 + krow]; aB.y = Bk[arow * 64 + krow + 1];
        o = __builtin_amdgcn_wmma_f32_16x16x4_f32(false, aR, false, bX,
                                                  (short)0, o, false, false);
        o = __builtin_amdgcn_wmma_f32_16x16x4_f32(false, aL, false, bV,
                                                  (short)0, o, false, false);
        x = __builtin_amdgcn_wmma_f32_16x16x4_f32(false, aM, false, bX,
                                                  (short)0, x, false, false);
        x = __builtin_amdgcn_wmma_f32_16x16x4_f32(false, aB, false, bV,
                                                  (short)0, x, false, false);
      }
      accO[mt] = o;
      accX[mt] = x;
    }

    // stream chunk outputs to HBM
    #pragma unroll
    for (int mt = 0; mt < 4; ++mt)
      #pragma unroll
      for (int r = 0; r < 8; ++r)
        out[(size_t)(k * CHUNK + mt * 16 + r + 8 * hi) * WW + col0 + lo] =
            accO[mt][r];

    // commit new state slice (wave-private LDS; DS ops in-order per wave)
    #pragma unroll
    for (int mt = 0; mt < 4; ++mt)
      #pragma unroll
      for (int r = 0; r < 8; ++r)
        XL[(mt * 16 + r + 8 * hi) * 16 + lo] = accX[mt][r];
  }
}

// ---------------------------------------------------------------------------
extern "C" void kernel_launch(void* const* d_in, const int* in_sizes, int n_in,
                              void* d_out, int out_size, void* d_ws,
                              size_t ws_size, hipStream_t stream) {
  (void)in_sizes; (void)n_in; (void)out_size; (void)ws_size;
  const float* actions = (const float*)d_in[0];   // [4096,3]
  const float* values  = (const float*)d_in[1];   // [4096,512]
  float* out = (float*)d_out;

  char* ws = (char*)d_ws;
  float* rw = (float*)ws; ws += (size_t)SEQ_LEN * 64 * sizeof(float);   // 1 MB
  float* Rg = (float*)ws; ws += (size_t)NCHUNK * 64 * 64 * sizeof(float);
  float* Lg = (float*)ws; ws += (size_t)NCHUNK * 64 * 64 * sizeof(float);
  float* Mg = (float*)ws; ws += (size_t)NCHUNK * 64 * 64 * sizeof(float);
  float* Bg = (float*)ws; ws += (size_t)NCHUNK * 64 * 64 * sizeof(float);

  ns_strength_scan<<<1, 32, 0, stream>>>(actions, rw);
  ns_chunk_mats<<<NCHUNK, 64, 0, stream>>>(actions, rw, Rg, Lg, Mg, Bg);
  ns_scan_gemm<<<4, 256, 0, stream>>>(values, Rg, Lg, Mg, Bg, out);
}